// SM_VFE_15444702396594
// MI455X (gfx1250) — compile-verified
//
#include <hip/hip_runtime.h>

typedef __attribute__((ext_vector_type(2))) float v2f;
typedef __attribute__((ext_vector_type(8))) float v8f;

#define NSWEEPS 10
#define C_IN 4
#define H 16
#define CATW (H * (NSWEEPS - 1)) // 144

static __device__ __forceinline__ v8f wmma4f32(v2f a, v2f b, v8f c) {
    // D = A(16x4 f32) * B(4x16 f32) + C(16x16 f32)
    return __builtin_amdgcn_wmma_f32_16x16x4_f32(
        /*neg_a=*/false, a, /*neg_b=*/false, b,
        /*c_mod=*/(short)0, c, /*reuse_a=*/false, /*reuse_b=*/false);
}

// raw v_rcp_f32 (1 trans op) instead of the IEEE div_scale/div_fmas chain
static __device__ __forceinline__ float fast_rcp(float x) {
    return __builtin_amdgcn_rcpf(x);
}

// sigmoid(x) = 0.5*tanh(0.5x) + 0.5 ; gfx1250 has hardware v_tanh_f32
static __device__ __forceinline__ float fast_sigmoid(float x) {
#if __has_builtin(__builtin_amdgcn_tanhf)
    return __builtin_fmaf(0.5f, __builtin_amdgcn_tanhf(0.5f * x), 0.5f);
#elif __has_builtin(__builtin_amdgcn_tanh_f32)
    return __builtin_fmaf(0.5f, __builtin_amdgcn_tanh_f32(0.5f * x), 0.5f);
#else
    return fast_rcp(1.f + __expf(-x));
#endif
}

__global__ __launch_bounds__(32) void vfe_wmma_kernel(
    const float* __restrict__ features,   // V x 10 x 4
    const float* __restrict__ W1,         // 4 x 16
    const float* __restrict__ b1,         // 16
    const float* __restrict__ W2,         // 16 x 16
    const float* __restrict__ b2,         // 16
    const float* __restrict__ W3,         // 144 x 16
    const float* __restrict__ b3,         // 16
    const int*   __restrict__ num_voxels, // V
    float* __restrict__ out,              // V x 20
    int V)
{
    // lds_diff: row-major by GEMM row rg = voxel*9 + sweep, 4 floats per row
    __shared__ float lds_diff[32 * 36];    // == 288 rows x 4
    __shared__ float lds_mot[16 * 16];     // one 16x16 D-tile staging buffer
    // lds_cat: row-major by rg, 16 floats per row; byte-identical to
    // [voxel][sweep*16+ch] since rg*16 = voxel*144 + sweep*16
    __shared__ float lds_cat[32 * CATW];

    const int lane  = threadIdx.x;            // 0..31
    const int vbase = blockIdx.x * 32;
    const int N     = lane & 15;              // WMMA column (B/C/D layouts)
    const int koff  = (lane < 16) ? 0 : 2;    // K sub-offset for A/B layouts
    const int Mbase = (lane < 16) ? 0 : 8;    // D-layout row base

    // ---------- preload weights into WMMA B-layout registers ----------
    v2f bw1;
    bw1.x = W1[(koff + 0) * H + N];
    bw1.y = W1[(koff + 1) * H + N];
    v2f bw2[4];
#pragma unroll
    for (int k = 0; k < 4; ++k) {
        bw2[k].x = W2[(4 * k + koff + 0) * H + N];
        bw2[k].y = W2[(4 * k + koff + 1) * H + N];
    }
    const float b1n = b1[N], b2n = b2[N], b3n = b3[N];

    // ---------- stage A: sweep means -> diff (to LDS), pm tail (to out) ----
    {
        const int vv = vbase + lane;
        const int v  = (vv < V) ? vv : (V - 1);
        const float4* fp = (const float4*)(features + (size_t)v * (NSWEEPS * C_IN));
        float4 pt[NSWEEPS];
        int    bkt[NSWEEPS];
        float sx = 0.f, sy = 0.f, sz = 0.f, st = 0.f;
#pragma unroll
        for (int p = 0; p < NSWEEPS; ++p) {
            pt[p] = fp[p];
            sx += pt[p].x; sy += pt[p].y; sz += pt[p].z; st += pt[p].w;
            // classify point p into its (unique) sweep bucket, or -1
            const float ts = pt[p].w;
            const bool pad = (pt[p].x == 0.f) && (pt[p].y == 0.f) &&
                             (pt[p].z == 0.f) && (pt[p].w == 0.f);
            const int   j  = (int)floorf(ts * 20.f + 0.1f);
            const float cj = 0.05f * (float)j;
            const bool inband = (j >= 1) && (j <= 8) &&
                                (ts >= cj - 0.005f) && (ts < cj + 0.005f);
            int b = -1;
            if ((ts == 0.f) && !pad) b = 0;
            if (inband)              b = j;
            if (ts > 0.445f)         b = NSWEEPS - 1;
            bkt[p] = b;
        }
        // per-sweep masked means; diff rows go to rg = lane*9 + (s-1)
        float m0x = 0.f, m0y = 0.f, m0z = 0.f, m0w = 0.f;
#pragma unroll
        for (int s = 0; s < NSWEEPS; ++s) {
            float mx = 0.f, my = 0.f, mz = 0.f, mw = 0.f, cc = 0.f;
#pragma unroll
            for (int p = 0; p < NSWEEPS; ++p) {
                const float m = (bkt[p] == s) ? 1.f : 0.f;
                mx += m * pt[p].x; my += m * pt[p].y;
                mz += m * pt[p].z; mw += m * pt[p].w; cc += m;
            }
            const float ic = fast_rcp(fmaxf(cc, 1.f));
            if (s == 0) {
                m0x = mx * ic; m0y = my * ic; m0z = mz * ic; m0w = mw * ic;
            } else {
                float* d = &lds_diff[(lane * 9 + (s - 1)) * 4];
                d[0] = m0x - mx * ic;
                d[1] = m0y - my * ic;
                d[2] = m0z - mz * ic;
                d[3] = m0w - mw * ic;
            }
        }
        const float inv = fast_rcp((float)num_voxels[v]);
        if (vv < V) {
            float* o = out + (size_t)vv * 20 + 16;
            o[0] = sx * inv; o[1] = sy * inv; o[2] = sz * inv; o[3] = st * inv;
        }
    }
    __syncthreads(); // single-wave WG: lowers to S_NOP

    // ---------- stages B + C: 288 rows = 18 tiles of (16 rows x 16 cols) ----
#pragma unroll 1
    for (int t = 0; t < 18; ++t) {
        // A-layout read of diff: lane holds row (t*16 + N), K = koff..koff+1
        // address is linear in row: 4*row + koff  (no div/mod needed)
        const int row = t * 16 + N;
        const v2f a = *(const v2f*)&lds_diff[row * 4 + koff];
        v8f acc = {};
        acc = wmma4f32(a, bw1, acc);

        // D-layout bias + relu -> mot, stage to LDS as plain 16x16 tile
        float mot[8];
#pragma unroll
        for (int r = 0; r < 8; ++r) {
            mot[r] = fmaxf(acc[r] + b1n, 0.f);
            lds_mot[(Mbase + r) * 16 + N] = mot[r];
        }
        __syncthreads();

        // stage C: K=16 as 4 chunks of K=4; A re-read transposed from LDS
        v8f z = {};
#pragma unroll
        for (int k = 0; k < 4; ++k) {
            const v2f am = *(const v2f*)&lds_mot[N * 16 + 4 * k + koff];
            z = wmma4f32(am, bw2[k], z);
        }

        // gated = sigmoid(z + b2) * mot; cat address is linear in rg:
        // gvl*144 + gs*16 + N == rg*16 + N  ->  t*256 + (Mbase+r)*16 + N
        float* catp = &lds_cat[t * 256 + Mbase * 16 + N];
#pragma unroll
        for (int r = 0; r < 8; ++r) {
            catp[r * 16] = mot[r] * fast_sigmoid(z[r] + b2n);
        }
        __syncthreads();
    }

    // ---------- stage D: temp = relu(cat @ W3 + b3), K = 144 = 36 x 4 ------
    v8f acc0 = {}, acc1 = {};
#pragma unroll 1
    for (int c = 0; c < 36; ++c) {
        const int kb = 4 * c + koff;
        v2f b;
        b.x = W3[(kb + 0) * H + N];
        b.y = W3[(kb + 1) * H + N];
        const v2f a0 = *(const v2f*)&lds_cat[(0 * 16 + N) * CATW + kb];
        const v2f a1 = *(const v2f*)&lds_cat[(1 * 16 + N) * CATW + kb];
        acc0 = wmma4f32(a0, b, acc0); // voxels vbase+0..15
        acc1 = wmma4f32(a1, b, acc1); // voxels vbase+16..31
    }
#pragma unroll
    for (int r = 0; r < 8; ++r) {
        const int v0 = vbase + Mbase + r;
        if (v0 < V)      out[(size_t)v0 * 20 + N] = fmaxf(acc0[r] + b3n, 0.f);
        const int v1 = vbase + 16 + Mbase + r;
        if (v1 < V)      out[(size_t)v1 * 20 + N] = fmaxf(acc1[r] + b3n, 0.f);
    }
}

extern "C" void kernel_launch(void* const* d_in, const int* in_sizes, int n_in,
                              void* d_out, int out_size, void* d_ws, size_t ws_size,
                              hipStream_t stream) {
    const float* features   = (const float*)d_in[0];
    const float* W1         = (const float*)d_in[1];
    const float* b1         = (const float*)d_in[2];
    const float* W2         = (const float*)d_in[3];
    const float* b2         = (const float*)d_in[4];
    const float* W3         = (const float*)d_in[5];
    const float* b3         = (const float*)d_in[6];
    const int*   num_voxels = (const int*)d_in[7];

    const int V = in_sizes[0] / (NSWEEPS * C_IN);
    const int blocks = (V + 31) / 32;
    vfe_wmma_kernel<<<blocks, 32, 0, stream>>>(
        features, W1, b1, W2, b2, W3, b3, num_voxels, (float*)d_out, V);
}